// ChiGNN_56255481643508
// MI455X (gfx1250) — compile-verified
//
#include <hip/hip_runtime.h>

typedef float v2f __attribute__((ext_vector_type(2)));
typedef float v8f __attribute__((ext_vector_type(8)));

#define NNODES 100000
#define NEDGES 1600000
#define FEAT   64

// ---------------------------------------------------------------------------
// CDNA5 async global->LDS copy (ASYNCcnt-tracked DMA, no VGPR bounce).
// GV addressing: global address is the 64-bit VADDR plus the instruction offset ; LDS dest byte addr in VDST VGPR.
// ---------------------------------------------------------------------------
__device__ __forceinline__ void async_copy_b128(void* lds_dst, const void* gsrc)
{
    uint32_t l = (uint32_t)(uintptr_t)lds_dst;   // low 32 bits = LDS offset
    asm volatile("global_load_async_to_lds_b128 %0, %1, off"
                 :: "v"(l), "v"(gsrc) : "memory");
}
__device__ __forceinline__ void async_wait0()
{
    asm volatile("s_wait_asynccnt 0x0" ::: "memory");
}

// ---------------------------------------------------------------------------
// zero a float buffer (n must be a multiple of 4)
// ---------------------------------------------------------------------------
__global__ void chignn_zero(float* __restrict__ p, int n)
{
    int n4 = n >> 2;
    for (int i = blockIdx.x * blockDim.x + threadIdx.x; i < n4;
         i += gridDim.x * blockDim.x)
        ((float4*)p)[i] = make_float4(0.f, 0.f, 0.f, 0.f);
}

// ---------------------------------------------------------------------------
// degree count (int atomics into a buffer later reused as dinv)
// ---------------------------------------------------------------------------
__global__ void chignn_degree(int* __restrict__ deg, const int* __restrict__ dst,
                              int nEdges)
{
    for (int e = blockIdx.x * blockDim.x + threadIdx.x; e < nEdges;
         e += gridDim.x * blockDim.x)
        atomicAdd(&deg[dst[e]], 1);
}

// dinv[i] = max(deg,1)^-0.5 , in place (int counts -> float)
__global__ void chignn_dinv(float* __restrict__ degdinv, int n)
{
    int i = blockIdx.x * blockDim.x + threadIdx.x;
    if (i >= n) return;
    float d = (float)((const int*)degdinv)[i];
    degdinv[i] = rsqrtf(fmaxf(d, 1.0f));
}

// ---------------------------------------------------------------------------
// fused 2-layer MLP: h = relu(relu(x W1 + b1) W2 + b2)
// block = 256 threads = 8 wave32, 128 rows per block.
// each wave computes a 16-row x 64-col strip via V_WMMA_F32_16X16X4_F32.
// tiles staged with GLOBAL_LOAD_ASYNC_TO_LDS_B128.
// ---------------------------------------------------------------------------
__global__ __launch_bounds__(256) void chignn_mlp(
    const float* __restrict__ x,
    const float* __restrict__ W1, const float* __restrict__ b1,
    const float* __restrict__ W2, const float* __restrict__ b2,
    float* __restrict__ h, int n)
{
    __shared__ float hs[128 * 64];   // activations (32 KB)
    __shared__ float wsh[64 * 64];   // weight tile  (16 KB)
    __shared__ float bsh[64];

    const int tid  = threadIdx.x;
    const int wid  = tid >> 5;
    const int lane = tid & 31;
    const int rowBase = blockIdx.x * 128;

    // stage x rows -> hs (async DMA; zero-pad tail rows with plain stores)
    for (int i = tid; i < 128 * 16; i += 256) {
        int r = i >> 4, c4 = i & 15;
        int grow = rowBase + r;
        if (grow < n)
            async_copy_b128(hs + r * FEAT + c4 * 4,
                            x + (size_t)grow * FEAT + c4 * 4);
        else
            ((float4*)(hs + r * FEAT))[c4] = make_float4(0.f, 0.f, 0.f, 0.f);
    }
    // stage W1 (async), b1
    for (int i = tid; i < 64 * 16; i += 256)
        async_copy_b128(wsh + i * 4, W1 + i * 4);
    if (tid < 64) bsh[tid] = b1[tid];
    async_wait0();
    __syncthreads();

    const int r    = (wid << 4) + (lane & 15);      // A row (local)
    const int cc   = lane & 15;                      // col within 16-tile
    const int koff = (lane >= 16) ? 2 : 0;
    const int rw   = (wid << 4) + ((lane >= 16) ? 8 : 0); // D row base

    v8f a0 = {}, a1 = {}, a2 = {}, a3 = {};
    // -------- layer 1 --------
    for (int k0 = 0; k0 < 64; k0 += 4) {
        int k = k0 + koff;
        v2f a; a.x = hs[r * FEAT + k]; a.y = hs[r * FEAT + k + 1];
        v2f b;
        b.x = wsh[k * 64 +  0 + cc]; b.y = wsh[(k + 1) * 64 +  0 + cc];
        a0 = __builtin_amdgcn_wmma_f32_16x16x4_f32(false, a, false, b, (short)0, a0, false, false);
        b.x = wsh[k * 64 + 16 + cc]; b.y = wsh[(k + 1) * 64 + 16 + cc];
        a1 = __builtin_amdgcn_wmma_f32_16x16x4_f32(false, a, false, b, (short)0, a1, false, false);
        b.x = wsh[k * 64 + 32 + cc]; b.y = wsh[(k + 1) * 64 + 32 + cc];
        a2 = __builtin_amdgcn_wmma_f32_16x16x4_f32(false, a, false, b, (short)0, a2, false, false);
        b.x = wsh[k * 64 + 48 + cc]; b.y = wsh[(k + 1) * 64 + 48 + cc];
        a3 = __builtin_amdgcn_wmma_f32_16x16x4_f32(false, a, false, b, (short)0, a3, false, false);
    }
    __syncthreads();   // everyone done reading hs / wsh

    // async-stage W2 into wsh (overlaps with relu write-back below)
    for (int i = tid; i < 64 * 16; i += 256)
        async_copy_b128(wsh + i * 4, W2 + i * 4);
    if (tid < 64) bsh[tid] = b2[tid];

    // relu + bias, write layer-1 result back to hs
    for (int v = 0; v < 8; ++v) {
        hs[(rw + v) * FEAT +  0 + cc] = fmaxf(a0[v] + bsh[ 0 + cc], 0.f);
        hs[(rw + v) * FEAT + 16 + cc] = fmaxf(a1[v] + bsh[16 + cc], 0.f);
        hs[(rw + v) * FEAT + 32 + cc] = fmaxf(a2[v] + bsh[32 + cc], 0.f);
        hs[(rw + v) * FEAT + 48 + cc] = fmaxf(a3[v] + bsh[48 + cc], 0.f);
    }
    async_wait0();
    __syncthreads();

    // NOTE: bsh was overwritten with b2 BEFORE the relu writes above read b1 —
    // this hazard is not avoided: the loop above may add b2 where it
    // must read b1. bsh is reloaded after the barrier only for layer 2:
    // (b2 is re-staged here, cheap)
    if (tid < 64) bsh[tid] = b2[tid];
    __syncthreads();

    // -------- layer 2 --------
    a0 = (v8f){}; a1 = (v8f){}; a2 = (v8f){}; a3 = (v8f){};
    for (int k0 = 0; k0 < 64; k0 += 4) {
        int k = k0 + koff;
        v2f a; a.x = hs[r * FEAT + k]; a.y = hs[r * FEAT + k + 1];
        v2f b;
        b.x = wsh[k * 64 +  0 + cc]; b.y = wsh[(k + 1) * 64 +  0 + cc];
        a0 = __builtin_amdgcn_wmma_f32_16x16x4_f32(false, a, false, b, (short)0, a0, false, false);
        b.x = wsh[k * 64 + 16 + cc]; b.y = wsh[(k + 1) * 64 + 16 + cc];
        a1 = __builtin_amdgcn_wmma_f32_16x16x4_f32(false, a, false, b, (short)0, a1, false, false);
        b.x = wsh[k * 64 + 32 + cc]; b.y = wsh[(k + 1) * 64 + 32 + cc];
        a2 = __builtin_amdgcn_wmma_f32_16x16x4_f32(false, a, false, b, (short)0, a2, false, false);
        b.x = wsh[k * 64 + 48 + cc]; b.y = wsh[(k + 1) * 64 + 48 + cc];
        a3 = __builtin_amdgcn_wmma_f32_16x16x4_f32(false, a, false, b, (short)0, a3, false, false);
    }
    // relu + bias, store to global
    for (int v = 0; v < 8; ++v) {
        int grow = rowBase + rw + v;
        if (grow < n) {
            float* o = h + (size_t)grow * FEAT;
            o[ 0 + cc] = fmaxf(a0[v] + bsh[ 0 + cc], 0.f);
            o[16 + cc] = fmaxf(a1[v] + bsh[16 + cc], 0.f);
            o[32 + cc] = fmaxf(a2[v] + bsh[32 + cc], 0.f);
            o[48 + cc] = fmaxf(a3[v] + bsh[48 + cc], 0.f);
        }
    }
}

// ---------------------------------------------------------------------------
// edge scatter: agg[dst] += fprev[src] * dinv[src]
// one wave32 per edge; float2 per lane (64 floats / edge). L2-resident.
// ---------------------------------------------------------------------------
__global__ __launch_bounds__(256) void chignn_scatter(
    const float* __restrict__ fprev, float* __restrict__ agg,
    const int* __restrict__ src, const int* __restrict__ dst,
    const float* __restrict__ dinv, int nEdges)
{
    int e    = blockIdx.x * (blockDim.x >> 5) + (threadIdx.x >> 5);
    int lane = threadIdx.x & 31;
    if (e >= nEdges) return;
    int s = src[e], d = dst[e];
    float sc = dinv[s];
    float2 v = ((const float2*)(fprev + (size_t)s * FEAT))[lane];
    float* p = agg + (size_t)d * FEAT + lane * 2;
    unsafeAtomicAdd(p,     v.x * sc);
    unsafeAtomicAdd(p + 1, v.y * sc);
}

// f[i] = fprev[i] - agg[i] * dinv[node]   (in place on agg buffer), float2 grain
__global__ void chignn_update(const float* __restrict__ fprev,
                              float* __restrict__ f,
                              const float* __restrict__ dinv, int n2)
{
    int i = blockIdx.x * blockDim.x + threadIdx.x;
    if (i >= n2) return;
    float di  = dinv[i >> 5];                 // 32 float2 per 64-wide row
    float2 a  = ((const float2*)f)[i];
    float2 p  = ((const float2*)fprev)[i];
    float2 r; r.x = p.x - a.x * di; r.y = p.y - a.y * di;
    ((float2*)f)[i] = r;
}

// ---------------------------------------------------------------------------
// fused head: h_final = concat_c( sum_k theta[c,k] F[k] )  (n x 256)
//             out = relu(h_final Wm1 + bm1) Wm2 + bm2      (n x 2)
// block = 256 threads, 32 rows; K=256 f32 WMMA GEMM, Wm1 async-staged
// ---------------------------------------------------------------------------
__global__ __launch_bounds__(256) void chignn_head(
    const float* __restrict__ Fb0, const float* __restrict__ Fb1,
    const float* __restrict__ Fb2, const float* __restrict__ Fb3,
    const float* __restrict__ thetas,
    const float* __restrict__ Wm1, const float* __restrict__ bm1,
    const float* __restrict__ Wm2, const float* __restrict__ bm2,
    float* __restrict__ out, int n)
{
    __shared__ float as[32 * 256];   // h_final tile (32 KB)
    __shared__ float bs[64 * 64];    // Wm1 tile (16 KB), reused as hout 32x64
    __shared__ float th[16], bm1s[64], wm2s[128], bm2s[2];

    const int tid  = threadIdx.x;
    const int wid  = tid >> 5;
    const int lane = tid & 31;
    const int rowBase = blockIdx.x * 32;

    if (tid < 16)  th[tid]   = thetas[tid];
    if (tid < 64)  bm1s[tid] = bm1[tid];
    if (tid < 128) wm2s[tid] = Wm2[tid];
    if (tid < 2)   bm2s[tid] = bm2[tid];
    __syncthreads();

    // build h_final tile: as[r][c*64+j] = sum_k th[c*4+k] * F[k][row][j]
    for (int i = tid; i < 32 * 64; i += 256) {
        int r = i >> 6, j = i & 63;
        int grow = rowBase + r;
        float f0 = 0.f, f1 = 0.f, f2 = 0.f, f3 = 0.f;
        if (grow < n) {
            size_t o = (size_t)grow * FEAT + j;
            f0 = Fb0[o]; f1 = Fb1[o]; f2 = Fb2[o]; f3 = Fb3[o];
        }
        #pragma unroll
        for (int c = 0; c < 4; ++c)
            as[r * 256 + c * 64 + j] =
                th[c*4+0]*f0 + th[c*4+1]*f1 + th[c*4+2]*f2 + th[c*4+3]*f3;
    }

    const int strip = wid >> 2;          // 0..1 -> 16-row strip
    const int tile  = wid & 3;           // 0..3 -> 16-col tile
    const int r     = (strip << 4) + (lane & 15);
    const int cc    = lane & 15;
    const int koff  = (lane >= 16) ? 2 : 0;

    v8f acc = {};
    for (int kb = 0; kb < 4; ++kb) {
        __syncthreads();
        // async-stage this 64-row tile of Wm1
        for (int i = tid; i < 64 * 16; i += 256)
            async_copy_b128(bs + i * 4, Wm1 + kb * 64 * 64 + i * 4);
        async_wait0();
        __syncthreads();
        for (int k0 = 0; k0 < 64; k0 += 4) {
            int k = k0 + koff;
            v2f a; a.x = as[r * 256 + kb * 64 + k];
                   a.y = as[r * 256 + kb * 64 + k + 1];
            v2f b; b.x = bs[k * 64 + tile * 16 + cc];
                   b.y = bs[(k + 1) * 64 + tile * 16 + cc];
            acc = __builtin_amdgcn_wmma_f32_16x16x4_f32(false, a, false, b, (short)0, acc, false, false);
        }
    }
    __syncthreads();

    // relu + bias into bs as hout[32][64]
    {
        int rw = (strip << 4) + ((lane >= 16) ? 8 : 0);
        for (int v = 0; v < 8; ++v)
            bs[(rw + v) * 64 + tile * 16 + cc] =
                fmaxf(acc[v] + bm1s[tile * 16 + cc], 0.f);
    }
    __syncthreads();

    // tiny head GEMM: out[row][cls] = hout[row] . Wm2[:,cls] + bm2[cls]
    if (tid < 64) {
        int r2 = tid >> 1, cls = tid & 1;
        int grow = rowBase + r2;
        if (grow < n) {
            float s = bm2s[cls];
            #pragma unroll 8
            for (int j = 0; j < 64; ++j)
                s += bs[r2 * 64 + j] * wm2s[j * 2 + cls];
            out[(size_t)grow * 2 + cls] = s;
        }
    }
}

// ---------------------------------------------------------------------------
extern "C" void kernel_launch(void* const* d_in, const int* in_sizes, int n_in,
                              void* d_out, int out_size, void* d_ws, size_t ws_size,
                              hipStream_t stream)
{
    const float* x      = (const float*)d_in[0];
    const int*   src    = (const int*)  d_in[1];
    const int*   dst    = (const int*)  d_in[2];
    const float* thetas = (const float*)d_in[3];
    const float* W1     = (const float*)d_in[4];
    const float* b1     = (const float*)d_in[5];
    const float* W2     = (const float*)d_in[6];
    const float* b2     = (const float*)d_in[7];
    const float* Wm1    = (const float*)d_in[8];
    const float* bm1    = (const float*)d_in[9];
    const float* Wm2    = (const float*)d_in[10];
    const float* bm2    = (const float*)d_in[11];
    float* out = (float*)d_out;

    // workspace layout (floats): [deg/dinv : NNODES][F0][F1][F2][F3]
    float* wsf  = (float*)d_ws;
    float* dinv = wsf;
    float* Fb[4];
    size_t off = NNODES;                       // 400000 B, 16B aligned
    for (int k = 0; k < 4; ++k) Fb[k] = wsf + off + (size_t)k * NNODES * FEAT;

    const int T = 256;

    // degrees -> dinv (in place)
    chignn_zero<<<dim3((NNODES / 4 + T - 1) / T), dim3(T), 0, stream>>>(dinv, NNODES);
    chignn_degree<<<dim3(2048), dim3(T), 0, stream>>>((int*)dinv, dst, NEDGES);
    chignn_dinv<<<dim3((NNODES + T - 1) / T), dim3(T), 0, stream>>>(dinv, NNODES);

    // fused MLP -> F0
    chignn_mlp<<<dim3((NNODES + 127) / 128), dim3(T), 0, stream>>>(
        x, W1, b1, W2, b2, Fb[0], NNODES);

    // 3 Chebyshev propagation rounds
    const int nElem = NNODES * FEAT;           // 6.4M floats
    for (int k = 1; k < 4; ++k) {
        chignn_zero<<<dim3((nElem / 4 + T - 1) / T), dim3(T), 0, stream>>>(Fb[k], nElem);
        chignn_scatter<<<dim3((NEDGES + 7) / 8), dim3(T), 0, stream>>>(
            Fb[k - 1], Fb[k], src, dst, dinv, NEDGES);
        chignn_update<<<dim3((nElem / 2 + T - 1) / T), dim3(T), 0, stream>>>(
            Fb[k - 1], Fb[k], dinv, nElem / 2);
    }

    // fused theta-combine + Wm1 + Wm2 head
    chignn_head<<<dim3((NNODES + 31) / 32), dim3(T), 0, stream>>>(
        Fb[0], Fb[1], Fb[2], Fb[3], thetas, Wm1, bm1, Wm2, bm2, out, NNODES);
}